// Loss_factory_12429635355015
// MI455X (gfx1250) — compile-verified
//
#include <hip/hip_runtime.h>
#include <math.h>

typedef float v2f __attribute__((ext_vector_type(2)));
typedef float v8f __attribute__((ext_vector_type(8)));

#define B_SZ   64
#define K_CLS  4
#define C_DIM  256
#define N_BANK 8192
#define KN_TOT (K_CLS * N_BANK)      // 32768
#define KN_PB  64                    // kn entries per block
#define NBLK   (KN_TOT / KN_PB)      // 512 blocks
#define APAD   260                   // padded LDS row stride (floats); 260%64=4 -> conflict-free

// workspace layout (floats):
//   [0, 16384)            anchors (l2-normalized mean of indiv)       64*256
//   [16384, 17408)        per-block {pos,neg} inter partials          2*NBLK
//   [17408, 17536)        per-sample {negl, intra} partials           2*B_SZ

// ---------------------------------------------------------------------------
// Kernel A: anchors[b][c] = l2norm(mean_p(indiv[b][p][c]))
// ---------------------------------------------------------------------------
__global__ __launch_bounds__(256) void anchors_kernel(
    const float* __restrict__ indiv, float* __restrict__ anchors) {
  __shared__ float red[256];
  __shared__ float invn;
  const int b = blockIdx.x;
  const int c = threadIdx.x;
  const float* base = indiv + (size_t)b * 4 * C_DIM;
  const float m = 0.25f * (base[c] + base[C_DIM + c] + base[2 * C_DIM + c] +
                           base[3 * C_DIM + c]);
  red[c] = m * m;
  __syncthreads();
  for (int s = 128; s > 0; s >>= 1) {
    if (c < s) red[c] += red[c + s];
    __syncthreads();
  }
  if (c == 0) invn = 1.0f / fmaxf(sqrtf(red[0]), 1e-12f);
  __syncthreads();
  anchors[(size_t)b * C_DIM + c] = m * invn;
}

// ---------------------------------------------------------------------------
// Kernel B: fused mean -> norm -> (anchors x bank) GEMM via WMMA fp32 ->
//           exp(sim/TAU) -> pos/neg partial sums per block. HBM-bound stream.
// ---------------------------------------------------------------------------
__global__ __launch_bounds__(256) void cohort_inter_kernel(
    const float* __restrict__ bank, const float* __restrict__ anchors,
    const int* __restrict__ label, float* __restrict__ partials) {
  __shared__ float sA[B_SZ][APAD];    // anchors, padded
  __shared__ float sB[KN_PB][APAD];   // bank means, padded
  __shared__ float sNrm[KN_PB][4];
  __shared__ float sInv[KN_PB];
  __shared__ float sPos[B_SZ];
  __shared__ float sW[8][2];

  const int tid = threadIdx.x;
  const int kn0 = blockIdx.x * KN_PB;
  const int kcls = kn0 >> 13;  // / N_BANK (KN_PB divides N_BANK -> uniform)

  if (tid < B_SZ) sPos[tid] = (label[tid] == kcls) ? 1.0f : 0.0f;

  // stage all 64 anchors into LDS (vectorized, coalesced)
  for (int i = tid; i < B_SZ * (C_DIM / 4); i += 256) {
    const int b = i >> 6, c4 = i & 63;
    const float4 v = ((const float4*)(anchors + (size_t)b * C_DIM))[c4];
    float* dst = &sA[b][c4 * 4];
    dst[0] = v.x; dst[1] = v.y; dst[2] = v.z; dst[3] = v.w;
  }
  // stream bank tile, reduce 4 components -> mean rows in LDS
  for (int i = tid; i < KN_PB * (C_DIM / 4); i += 256) {
    const int e = i >> 6, c4 = i & 63;
    const float4* base = (const float4*)(bank + (size_t)(kn0 + e) * 4 * C_DIM);
    const float4 v0 = base[c4];
    const float4 v1 = base[64 + c4];
    const float4 v2 = base[128 + c4];
    const float4 v3 = base[192 + c4];
    float* dst = &sB[e][c4 * 4];
    dst[0] = 0.25f * (v0.x + v1.x + v2.x + v3.x);
    dst[1] = 0.25f * (v0.y + v1.y + v2.y + v3.y);
    dst[2] = 0.25f * (v0.z + v1.z + v2.z + v3.z);
    dst[3] = 0.25f * (v0.w + v1.w + v2.w + v3.w);
  }
  __syncthreads();

  // per-entry inverse norms (4 threads per entry)
  {
    const int e = tid >> 2, q = tid & 3;
    const float* row = &sB[e][q * 64];
    float ss = 0.0f;
    for (int c = 0; c < 64; ++c) ss += row[c] * row[c];
    sNrm[e][q] = ss;
  }
  __syncthreads();
  if (tid < KN_PB) {
    const float ss = sNrm[tid][0] + sNrm[tid][1] + sNrm[tid][2] + sNrm[tid][3];
    sInv[tid] = 1.0f / fmaxf(sqrtf(ss), 1e-12f);
  }
  __syncthreads();

  // WMMA: D(16b x 16kn) += A(16b x 4c) * B(4c x 16kn), fp32 throughout.
  const int wave = tid >> 5, lane = tid & 31;
  const int b0 = (wave & 3) * 16;   // anchor tile
  const int n0 = (wave >> 2) * 32;  // two kn tiles: n0, n0+16
  const int m = lane & 15, h = lane >> 4;

  v8f acc0 = {};
  v8f acc1 = {};
  const float* aRow = &sA[b0 + m][0];
  const float* bRow0 = &sB[n0 + m][0];
  const float* bRow1 = &sB[n0 + 16 + m][0];
  for (int c = 0; c < C_DIM; c += 4) {
    const int off = c + 2 * h;  // lanes 0-15 hold K{0,1}; lanes 16-31 hold K{2,3}
    v2f af;  af.x = aRow[off];  af.y = aRow[off + 1];
    v2f bf0; bf0.x = bRow0[off]; bf0.y = bRow0[off + 1];
    v2f bf1; bf1.x = bRow1[off]; bf1.y = bRow1[off + 1];
    acc0 = __builtin_amdgcn_wmma_f32_16x16x4_f32(false, af, false, bf0,
                                                 (short)0, acc0, false, false);
    acc1 = __builtin_amdgcn_wmma_f32_16x16x4_f32(false, af, false, bf1,
                                                 (short)0, acc1, false, false);
  }

  // epilogue: scale by inv-norm, exp(sim/TAU), split pos/neg by label
  float pos = 0.0f, neg = 0.0f;
  const int nd = lane & 15;
  const float inv0 = sInv[n0 + nd];
  const float inv1 = sInv[n0 + 16 + nd];
  for (int g = 0; g < 8; ++g) {  // D VGPR g: rows g (lanes 0-15) and g+8 (16-31)
    const int bg = b0 + g + 8 * h;
    const float pf = sPos[bg];
    const float e0 = expf(acc0[g] * inv0 * 0.5f);  // /TAU, TAU=2
    const float e1 = expf(acc1[g] * inv1 * 0.5f);
    pos += pf * (e0 + e1);
    neg += (1.0f - pf) * (e0 + e1);
  }
  // wave32 reduction
  for (int d = 16; d >= 1; d >>= 1) {
    pos += __shfl_xor(pos, d, 32);
    neg += __shfl_xor(neg, d, 32);
  }
  if (lane == 0) { sW[wave][0] = pos; sW[wave][1] = neg; }
  __syncthreads();
  if (tid == 0) {
    float tp = 0.0f, tn = 0.0f;
    for (int w = 0; w < 8; ++w) { tp += sW[w][0]; tn += sW[w][1]; }
    partials[2 * blockIdx.x] = tp;      // deterministic per-block partials
    partials[2 * blockIdx.x + 1] = tn;  // (no float atomics)
  }
}

// ---------------------------------------------------------------------------
// Kernel C1: per-sample NLL + intra loss, fully parallel & coalesced.
// Block b: thread c owns column c; 14 partial sums (4 indiv norms, gene/path
// norms, 8 raw dots) reduced via LDS trees; thread 0 applies mask algebra.
// ---------------------------------------------------------------------------
__global__ __launch_bounds__(256) void intra_nll_kernel(
    const float* __restrict__ hazards, const float* __restrict__ S,
    const float* __restrict__ indiv, const float* __restrict__ gene,
    const float* __restrict__ path, const int* __restrict__ label,
    const int* __restrict__ cens, float* __restrict__ bpart) {
  __shared__ float red[14][272];  // 272 stride: conflict-free tree reduction
  const int b = blockIdx.x;
  const int c = threadIdx.x;

  const float* iv = indiv + (size_t)b * 4 * C_DIM;
  const float a0 = iv[c];
  const float a1 = iv[C_DIM + c];
  const float a2 = iv[2 * C_DIM + c];
  const float a3 = iv[3 * C_DIM + c];
  const float g = gene[(size_t)b * C_DIM + c];
  const float p = path[(size_t)b * C_DIM + c];

  red[0][c] = a0 * a0;  red[1][c] = a1 * a1;
  red[2][c] = a2 * a2;  red[3][c] = a3 * a3;
  red[4][c] = g * g;    red[5][c] = p * p;
  red[6][c] = a0 * g;   red[7][c] = a0 * p;
  red[8][c] = a1 * g;   red[9][c] = a1 * p;
  red[10][c] = a2 * g;  red[11][c] = a2 * p;
  red[12][c] = a3 * g;  red[13][c] = a3 * p;
  __syncthreads();
  for (int s = 128; s > 0; s >>= 1) {
    if (c < s) {
#pragma unroll
      for (int j = 0; j < 14; ++j) red[j][c] += red[j][c + s];
    }
    __syncthreads();
  }

  if (c == 0) {
    float invI[4];
#pragma unroll
    for (int r = 0; r < 4; ++r)
      invI[r] = 1.0f / fmaxf(sqrtf(red[r][0]), 1e-12f);
    const float invG = 1.0f / fmaxf(sqrtf(red[4][0]), 1e-12f);
    const float invP = 1.0f / fmaxf(sqrtf(red[5][0]), 1e-12f);

    // sim[r][t]; mask rows: (1,1),(0,0),(1,0),(0,1)
    const float s00 = red[6][0] * invI[0] * invG;
    const float s01 = red[7][0] * invI[0] * invP;
    const float s10 = red[8][0] * invI[1] * invG;
    const float s11 = red[9][0] * invI[1] * invP;
    const float s20 = red[10][0] * invI[2] * invG;
    const float s21 = red[11][0] * invI[2] * invP;
    const float s30 = red[12][0] * invI[3] * invG;
    const float s31 = red[13][0] * invI[3] * invP;
    const float acc = (-s00) + (-s01)            // common: -sim
                      + fabsf(s10) + fabsf(s11)  // synergy: |sim|
                      + (-s20) + fabsf(s21)      // g_spec
                      + fabsf(s30) + (-s31);     // p_spec
    const float intra_b = acc * (1.0f / 8.0f) + 1.0f;

    // NLL (ALPHA=0)
    const int Y = label[b];
    const float cf = (float)cens[b];
    const float s_y = (Y == 0) ? 1.0f : S[b * K_CLS + Y - 1];  // S_padded[Y]
    const float s_y1 = S[b * K_CLS + Y];                       // S_padded[Y+1]
    const float h_y = hazards[b * K_CLS + Y];
    const float negl =
        -cf * logf(fmaxf(s_y1, 1e-7f)) -
        (1.0f - cf) * (logf(fmaxf(s_y, 1e-7f)) + logf(fmaxf(h_y, 1e-7f)));

    bpart[2 * b] = negl;
    bpart[2 * b + 1] = intra_b;
  }
}

// ---------------------------------------------------------------------------
// Kernel C2: combine all partials -> scalar loss
// ---------------------------------------------------------------------------
__global__ __launch_bounds__(64) void reduce_final_kernel(
    const float* __restrict__ partials, const float* __restrict__ bpart,
    float* __restrict__ out) {
  __shared__ float rp[64], rq[64], rn[64], ri[64];
  const int t = threadIdx.x;
  float tp = 0.0f, tn = 0.0f;
  for (int i = t; i < NBLK; i += 64) {
    tp += partials[2 * i];
    tn += partials[2 * i + 1];
  }
  rp[t] = tp;
  rq[t] = tn;
  rn[t] = bpart[2 * t];
  ri[t] = bpart[2 * t + 1];
  __syncthreads();
  if (t == 0) {
    float sp = 0.0f, sq = 0.0f, sn = 0.0f, si = 0.0f;
    for (int i = 0; i < 64; ++i) {
      sp += rp[i]; sq += rq[i]; sn += rn[i]; si += ri[i];
    }
    const float nll = sn * (1.0f / 64.0f);
    const float intra = si * (1.0f / 64.0f);
    const float ep = sp / (float)((size_t)B_SZ * N_BANK);
    const float en = sq / (float)((size_t)B_SZ * (K_CLS - 1) * N_BANK);
    const float inter = -logf((ep + 1e-8f) / (ep + en + 1e-8f));
    out[0] = nll + intra + inter;
  }
}

// ---------------------------------------------------------------------------
extern "C" void kernel_launch(void* const* d_in, const int* in_sizes, int n_in,
                              void* d_out, int out_size, void* d_ws,
                              size_t ws_size, hipStream_t stream) {
  (void)in_sizes; (void)n_in; (void)out_size; (void)ws_size;
  const float* hazards = (const float*)d_in[0];
  const float* S       = (const float*)d_in[1];
  const float* indiv   = (const float*)d_in[2];
  const float* gene    = (const float*)d_in[3];
  const float* path    = (const float*)d_in[4];
  const float* bank    = (const float*)d_in[5];
  const int*   label   = (const int*)d_in[6];
  const int*   cens    = (const int*)d_in[7];
  float* out = (float*)d_out;

  float* anchors  = (float*)d_ws;               // 64*256 floats
  float* partials = anchors + B_SZ * C_DIM;     // 2*NBLK floats
  float* bpart    = partials + 2 * NBLK;        // 2*B_SZ floats

  hipLaunchKernelGGL(anchors_kernel, dim3(B_SZ), dim3(256), 0, stream,
                     indiv, anchors);
  hipLaunchKernelGGL(cohort_inter_kernel, dim3(NBLK), dim3(256), 0, stream,
                     bank, anchors, label, partials);
  hipLaunchKernelGGL(intra_nll_kernel, dim3(B_SZ), dim3(256), 0, stream,
                     hazards, S, indiv, gene, path, label, cens, bpart);
  hipLaunchKernelGGL(reduce_final_kernel, dim3(1), dim3(64), 0, stream,
                     partials, bpart, out);
}